// TenHotEncodeLayer_26233660244449
// MI455X (gfx1250) — compile-verified
//
#include <hip/hip_runtime.h>
#include <stdint.h>

// out[n,t,x[n,t,f]] = 1.0, zeros elsewhere. N=32, T=512, F=10, TOKENS=5000.
// Pure store-bandwidth problem: 327.68 MB of output -> ~14 us at 23.3 TB/s.
// One block per (n,t) row: zero the row with async LDS->global B128 stores
// (CDNA5 async engine, ASYNCcnt), wait, barrier, then scatter the 10 ones.

#define NUM_TOKENS 5000
#define FEATS      10
#define THREADS    256
#define CHUNKS     (NUM_TOKENS / 4)   // 1250 float4 chunks per row (5000 % 4 == 0)

typedef __attribute__((ext_vector_type(4))) int v4i;
typedef __attribute__((address_space(1))) v4i g_v4i;   // global 128-bit chunk
typedef __attribute__((address_space(3))) v4i l_v4i;   // LDS 128-bit chunk

__device__ __forceinline__ void async_zero_store_b128(float* gptr, const void* lptr) {
#if __has_builtin(__builtin_amdgcn_global_store_async_from_lds_b128)
    __builtin_amdgcn_global_store_async_from_lds_b128(
        (g_v4i*)gptr, (l_v4i*)lptr, /*offset=*/0, /*cpol=*/0);
#else
    uint64_t ga = (uint64_t)(uintptr_t)gptr;
    uint32_t la = (uint32_t)(uintptr_t)lptr;   // low 32 bits of generic LDS addr = LDS offset
    asm volatile("global_store_async_from_lds_b128 %0, %1, off"
                 :: "v"(ga), "v"(la)
                 : "memory");
#endif
}

__device__ __forceinline__ void wait_asynccnt0() {
#if __has_builtin(__builtin_amdgcn_s_wait_asynccnt)
    __builtin_amdgcn_s_wait_asynccnt(0);
#else
    asm volatile("s_wait_asynccnt 0x0" ::: "memory");
#endif
}

__global__ __launch_bounds__(THREADS)
void TenHotEncode_kernel(const int* __restrict__ x,
                         float* __restrict__ out,
                         int rows) {
    __shared__ float4 zbuf[THREADS];          // 4 KB of zeros, source tile for async stores
    const int tid = threadIdx.x;
    const int row = blockIdx.x;
    if (row >= rows) return;                  // uniform per block; safe with barriers

    // Stage a 16B zero tile per lane in LDS.
    zbuf[tid] = make_float4(0.f, 0.f, 0.f, 0.f);
    __syncthreads();                          // LDS writes visible before async engine reads

    float* rowbase = out + (size_t)row * NUM_TOKENS;   // 20000 B per row, 16B-aligned

    // Stream zeros LDS -> global: 1250 x 16B chunks per row via async B128 stores.
    const void* lsrc = &zbuf[tid];
    for (int c = tid; c < CHUNKS; c += THREADS) {
        async_zero_store_b128(rowbase + (c << 2), lsrc);
    }

    // Async stores are unordered vs. normal VMEM stores: drain them, then
    // barrier so every wave's zeros landed before any lane writes a 1.0.
    wait_asynccnt0();
    __syncthreads();

    // Scatter the 10 ones for this row.
    if (tid < FEATS) {
        int idx = x[row * FEATS + tid];
        if ((unsigned)idx < (unsigned)NUM_TOKENS) {
            rowbase[idx] = 1.0f;
        }
    }
}

extern "C" void kernel_launch(void* const* d_in, const int* in_sizes, int n_in,
                              void* d_out, int out_size, void* d_ws, size_t ws_size,
                              hipStream_t stream) {
    const int* x = (const int*)d_in[0];       // [N, T, F] int32 indices
    float* out = (float*)d_out;               // [N, T, NUM_TOKENS] float32
    const int rows = in_sizes[0] / FEATS;     // N*T = 16384

    dim3 grid(rows);
    dim3 block(THREADS);
    hipLaunchKernelGGL(TenHotEncode_kernel, grid, block, 0, stream, x, out, rows);
}